// HeteroSAGE_27522150432791
// MI455X (gfx1250) — compile-verified
//
#include <hip/hip_runtime.h>

// CDNA5 / gfx1250 HeteroSAGE. fp32 WMMA (16x16x4) GEMMs + f32 global atomics
// for the segment-mean scatter. wave32 layout per CDNA5 ISA 7.12.2.

typedef __attribute__((ext_vector_type(2))) float v2f;
typedef __attribute__((ext_vector_type(8))) float v8f;

#define THREADS 256
#define MT 64  // M rows per block in the WMMA GEMM

// ---------------------------------------------------------------- counts
__global__ void k_count(const int* __restrict__ dst, long long E,
                        float* __restrict__ cnt) {
  long long e = (long long)blockIdx.x * blockDim.x + threadIdx.x;
  if (e < E) atomicAdd(&cnt[dst[e]], 1.0f);
}

__global__ void k_inv(float* __restrict__ c, int n) {
  int i = blockIdx.x * blockDim.x + threadIdx.x;
  if (i < n) c[i] = 1.0f / fmaxf(c[i], 1.0f);
}

// ---------------------------------------------------------------- scatter
// one thread = (edge, 4 consecutive channels). K = 4 << kshift2.
__global__ void k_scatter(const float* __restrict__ xsrc,
                          const int* __restrict__ src,
                          const int* __restrict__ dst,
                          long long E, int kshift2,
                          float* __restrict__ acc) {
  long long gid = (long long)blockIdx.x * blockDim.x + threadIdx.x;
  long long tot = E << kshift2;
  if (gid >= tot) return;
  long long e = gid >> kshift2;
  int c4 = ((int)gid & ((1 << kshift2) - 1)) << 2;
  int K = 4 << kshift2;
  int s = src[e], d = dst[e];
  const float4 v = *(const float4*)(xsrc + (long long)s * K + c4);
  float* a = acc + (long long)d * K + c4;
  atomicAdd(a + 0, v.x);
  atomicAdd(a + 1, v.y);
  atomicAdd(a + 2, v.z);
  atomicAdd(a + 3, v.w);
}

// ---------------------------------------------------------------- weight packing
// Wp row-major [128 x 2K] = [Wl | Wr]
__global__ void k_pack2(const float* __restrict__ Wl, const float* __restrict__ Wr,
                        int K, float* __restrict__ Wp) {
  int Ktot = 2 * K;
  int i = blockIdx.x * blockDim.x + threadIdx.x;
  if (i >= 128 * Ktot) return;
  int j = i / Ktot, k = i % Ktot;
  Wp[i] = (k < K) ? Wl[j * K + k] : Wr[j * K + (k - K)];
}

// Wp row-major [128 x 3K] = [WlA | WlB | WrA+WrB]
__global__ void k_pack3(const float* __restrict__ WlA, const float* __restrict__ WlB,
                        const float* __restrict__ WrA, const float* __restrict__ WrB,
                        int K, float* __restrict__ Wp) {
  int Ktot = 3 * K;
  int i = blockIdx.x * blockDim.x + threadIdx.x;
  if (i >= 128 * Ktot) return;
  int j = i / Ktot, k = i % Ktot;
  float v;
  if (k < K) v = WlA[j * K + k];
  else if (k < 2 * K) v = WlB[j * K + (k - K)];
  else v = WrA[j * K + (k - 2 * K)] + WrB[j * K + (k - 2 * K)];
  Wp[i] = v;
}

__global__ void k_bias2(const float* __restrict__ a, const float* __restrict__ b,
                        float* __restrict__ o) {
  int j = threadIdx.x;
  if (j < 128) o[j] = a[j] + (b ? b[j] : 0.0f);
}

// ---------------------------------------------------------------- WMMA GEMM
// out[n, 0..127] = epilogue( sum_b  Ab[n,:]*invb[n] @ Wp_b^T  + bias )
// 8 waves: wave w -> cols [16w,16w+16).  MT=64 rows/block, 4 M-subtiles/wave.
__global__ __launch_bounds__(THREADS)
void k_sage_gemm(const float* __restrict__ X0, const float* __restrict__ X1,
                 const float* __restrict__ X2,
                 const float* __restrict__ inv0, const float* __restrict__ inv1,
                 const float* __restrict__ inv2,
                 int nb, int kshift, int N, int Ktot,
                 const float* __restrict__ Wp, const float* __restrict__ bias,
                 float scale, int do_relu,
                 const float* __restrict__ addin, float* __restrict__ out) {
  __shared__ float sA[MT * 128];  // 32 KB max (K<=128)
  const int K = 1 << kshift;
  const int tid = threadIdx.x;
  const int lane = tid & 31, wave = tid >> 5;
  const int llo = lane & 15, lhi = lane >> 4;
  const int row0 = blockIdx.x * MT;

  v8f c0 = {}, c1 = {}, c2 = {}, c3 = {};

  for (int b = 0; b < nb; ++b) {
    const float* X = (b == 0) ? X0 : ((b == 1) ? X1 : X2);
    const float* inv = (b == 0) ? inv0 : ((b == 1) ? inv1 : inv2);

    __syncthreads();  // sA reuse (no-op hazard for b==0)
    for (int i = tid; i < MT * K; i += THREADS) {
      int r = i >> kshift, k = i & (K - 1);
      int g = row0 + r;
      float v = 0.0f;
      if (g < N) {
        v = X[(long long)g * K + k];
        if (inv) v *= inv[g];
      }
      sA[(r << kshift) + k] = v;
    }
    __syncthreads();

    // B fragment base: lane holds W[16w+llo][bK + ks + 2*lhi + {0,1}]
    const float* Wb = Wp + (long long)(wave * 16 + llo) * Ktot + b * K + (lhi << 1);
    const float* arow = &sA[(llo << kshift) + (lhi << 1)];

    for (int ks = 0; ks < K; ks += 4) {
      v2f bf = *(const v2f*)(Wb + ks);
      const float* ap = arow + ks;
      v2f a0 = *(const v2f*)(ap);
      v2f a1 = *(const v2f*)(ap + (16 << kshift));
      v2f a2 = *(const v2f*)(ap + (32 << kshift));
      v2f a3 = *(const v2f*)(ap + (48 << kshift));
      c0 = __builtin_amdgcn_wmma_f32_16x16x4_f32(false, a0, false, bf, (short)0, c0, false, false);
      c1 = __builtin_amdgcn_wmma_f32_16x16x4_f32(false, a1, false, bf, (short)0, c1, false, false);
      c2 = __builtin_amdgcn_wmma_f32_16x16x4_f32(false, a2, false, bf, (short)0, c2, false, false);
      c3 = __builtin_amdgcn_wmma_f32_16x16x4_f32(false, a3, false, bf, (short)0, c3, false, false);
    }
  }

  // epilogue: D[m*16 + v + 8*lhi][16w + llo]
  const int col = wave * 16 + llo;
  const float bv = bias[col];
  v8f cs[4] = {c0, c1, c2, c3};
#pragma unroll
  for (int m = 0; m < 4; ++m) {
#pragma unroll
    for (int vv = 0; vv < 8; ++vv) {
      int r = row0 + m * 16 + vv + (lhi << 3);
      if (r < N) {
        float val = scale * (cs[m][vv] + bv);
        if (do_relu) val = fmaxf(val, 0.0f);
        if (addin) val = 0.5f * (addin[(long long)r * 128 + col] + val);
        out[(long long)r * 128 + col] = val;
      }
    }
  }
}

// ---------------------------------------------------------------- host
static inline unsigned gdiv(long long a, int b) { return (unsigned)((a + b - 1) / b); }

extern "C" void kernel_launch(void* const* d_in, const int* in_sizes, int n_in,
                              void* d_out, int out_size, void* d_ws, size_t ws_size,
                              hipStream_t stream) {
  const long long NU = 100000, NR = 200000, NI = 10000;
  const long long ER = 2000000, EH = 1000000;

  const float* x_user   = (const float*)d_in[0];
  const float* x_recipe = (const float*)d_in[1];
  const float* x_ing    = (const float*)d_in[2];
  const int* rs = (const int*)d_in[3];   // rates_src (user)
  const int* rd = (const int*)d_in[4];   // rates_dst (recipe)
  const int* hs = (const int*)d_in[5];   // has_src (recipe)
  const int* hd = (const int*)d_in[6];   // has_dst (ing)
  // params: 7 + layer*12 + rel*3 + {0:Wl,1:b,2:Wr}; rel: 0=ur 1=ru 2=ri 3=ir
  auto P = [&](int layer, int rel, int which) -> const float* {
    return (const float*)d_in[7 + layer * 12 + rel * 3 + which];
  };

  // ---- workspace carve (256B aligned)
  char* w = (char*)d_ws;
  auto alloc = [&](size_t bytes) -> float* {
    float* p = (float*)w;
    w += (bytes + 255) & ~(size_t)255;
    return p;
  };
  float* inv_ur = alloc(NR * 4);
  float* inv_ru = alloc(NU * 4);
  float* inv_ri = alloc(NI * 4);
  float* inv_ir = alloc(NR * 4);
  float* h1u = alloc(NU * 128 * 4);
  float* h1r = alloc(NR * 128 * 4);
  float* h1i = alloc(NI * 128 * 4);
  float* accbase = alloc((NR + NU + NI + NR) * 128 * 4);  // reused across layers
  float* Wp1r = alloc(128 * 192 * 4);
  float* Wp1u = alloc(128 * 128 * 4);
  float* Wp1i = alloc(128 * 128 * 4);
  float* Wp2r = alloc(128 * 384 * 4);
  float* Wp2u = alloc(128 * 256 * 4);
  float* Wp2i = alloc(128 * 256 * 4);
  float* b1r = alloc(128 * 4); float* b1u = alloc(128 * 4); float* b1i = alloc(128 * 4);
  float* b2r = alloc(128 * 4); float* b2u = alloc(128 * 4); float* b2i = alloc(128 * 4);

  // ---- degree counts (shared by both layers)
  hipMemsetAsync(inv_ur, 0, NR * 4, stream);
  hipMemsetAsync(inv_ru, 0, NU * 4, stream);
  hipMemsetAsync(inv_ri, 0, NI * 4, stream);
  hipMemsetAsync(inv_ir, 0, NR * 4, stream);
  k_count<<<gdiv(ER, THREADS), THREADS, 0, stream>>>(rd, ER, inv_ur);
  k_count<<<gdiv(ER, THREADS), THREADS, 0, stream>>>(rs, ER, inv_ru);
  k_count<<<gdiv(EH, THREADS), THREADS, 0, stream>>>(hd, EH, inv_ri);
  k_count<<<gdiv(EH, THREADS), THREADS, 0, stream>>>(hs, EH, inv_ir);
  k_inv<<<gdiv(NR, THREADS), THREADS, 0, stream>>>(inv_ur, (int)NR);
  k_inv<<<gdiv(NU, THREADS), THREADS, 0, stream>>>(inv_ru, (int)NU);
  k_inv<<<gdiv(NI, THREADS), THREADS, 0, stream>>>(inv_ri, (int)NI);
  k_inv<<<gdiv(NR, THREADS), THREADS, 0, stream>>>(inv_ir, (int)NR);

  // ================= layer 1 (K=64) =================
  {
    const int K = 64, cks = 4;  // chunks = K/4 = 16
    float* aUr = accbase;
    float* aRu = aUr + NR * K;
    float* aRi = aRu + NU * K;
    float* aIr = aRi + NI * K;
    hipMemsetAsync(accbase, 0, (size_t)(NR + NU + NI + NR) * K * 4, stream);
    k_scatter<<<gdiv(ER << cks, THREADS), THREADS, 0, stream>>>(x_user,   rs, rd, ER, cks, aUr);
    k_scatter<<<gdiv(ER << cks, THREADS), THREADS, 0, stream>>>(x_recipe, rd, rs, ER, cks, aRu);
    k_scatter<<<gdiv(EH << cks, THREADS), THREADS, 0, stream>>>(x_recipe, hs, hd, EH, cks, aRi);
    k_scatter<<<gdiv(EH << cks, THREADS), THREADS, 0, stream>>>(x_ing,    hd, hs, EH, cks, aIr);

    k_pack3<<<gdiv(128 * 3 * K, THREADS), THREADS, 0, stream>>>(
        P(0,0,0), P(0,3,0), P(0,0,2), P(0,3,2), K, Wp1r);
    k_pack2<<<gdiv(128 * 2 * K, THREADS), THREADS, 0, stream>>>(P(0,1,0), P(0,1,2), K, Wp1u);
    k_pack2<<<gdiv(128 * 2 * K, THREADS), THREADS, 0, stream>>>(P(0,2,0), P(0,2,2), K, Wp1i);
    k_bias2<<<1, 128, 0, stream>>>(P(0,0,1), P(0,3,1), b1r);
    k_bias2<<<1, 128, 0, stream>>>(P(0,1,1), nullptr,  b1u);
    k_bias2<<<1, 128, 0, stream>>>(P(0,2,1), nullptr,  b1i);

    k_sage_gemm<<<gdiv(NR, MT), THREADS, 0, stream>>>(
        aUr, aIr, x_recipe, inv_ur, inv_ir, nullptr,
        3, 6, (int)NR, 3 * K, Wp1r, b1r, 0.5f, 1, nullptr, h1r);
    k_sage_gemm<<<gdiv(NU, MT), THREADS, 0, stream>>>(
        aRu, x_user, nullptr, inv_ru, nullptr, nullptr,
        2, 6, (int)NU, 2 * K, Wp1u, b1u, 1.0f, 1, nullptr, h1u);
    k_sage_gemm<<<gdiv(NI, MT), THREADS, 0, stream>>>(
        aRi, x_ing, nullptr, inv_ri, nullptr, nullptr,
        2, 6, (int)NI, 2 * K, Wp1i, b1i, 1.0f, 1, nullptr, h1i);
  }

  // ================= layer 2 (K=128), fused (h1+h2)*0.5 into d_out =====
  {
    const int K = 128, cks = 5;  // chunks = K/4 = 32
    float* aUr = accbase;
    float* aRu = aUr + NR * K;
    float* aRi = aRu + NU * K;
    float* aIr = aRi + NI * K;
    hipMemsetAsync(accbase, 0, (size_t)(NR + NU + NI + NR) * K * 4, stream);
    k_scatter<<<gdiv(ER << cks, THREADS), THREADS, 0, stream>>>(h1u, rs, rd, ER, cks, aUr);
    k_scatter<<<gdiv(ER << cks, THREADS), THREADS, 0, stream>>>(h1r, rd, rs, ER, cks, aRu);
    k_scatter<<<gdiv(EH << cks, THREADS), THREADS, 0, stream>>>(h1r, hs, hd, EH, cks, aRi);
    k_scatter<<<gdiv(EH << cks, THREADS), THREADS, 0, stream>>>(h1i, hd, hs, EH, cks, aIr);

    k_pack3<<<gdiv(128 * 3 * K, THREADS), THREADS, 0, stream>>>(
        P(1,0,0), P(1,3,0), P(1,0,2), P(1,3,2), K, Wp2r);
    k_pack2<<<gdiv(128 * 2 * K, THREADS), THREADS, 0, stream>>>(P(1,1,0), P(1,1,2), K, Wp2u);
    k_pack2<<<gdiv(128 * 2 * K, THREADS), THREADS, 0, stream>>>(P(1,2,0), P(1,2,2), K, Wp2i);
    k_bias2<<<1, 128, 0, stream>>>(P(1,0,1), P(1,3,1), b2r);
    k_bias2<<<1, 128, 0, stream>>>(P(1,1,1), nullptr,  b2u);
    k_bias2<<<1, 128, 0, stream>>>(P(1,2,1), nullptr,  b2i);

    float* outU = (float*)d_out;
    float* outR = outU + NU * 128;
    float* outI = outR + NR * 128;
    k_sage_gemm<<<gdiv(NR, MT), THREADS, 0, stream>>>(
        aUr, aIr, h1r, inv_ur, inv_ir, nullptr,
        3, 7, (int)NR, 3 * K, Wp2r, b2r, 0.5f, 0, h1r, outR);
    k_sage_gemm<<<gdiv(NU, MT), THREADS, 0, stream>>>(
        aRu, h1u, nullptr, inv_ru, nullptr, nullptr,
        2, 7, (int)NU, 2 * K, Wp2u, b2u, 1.0f, 0, h1u, outU);
    k_sage_gemm<<<gdiv(NI, MT), THREADS, 0, stream>>>(
        aRi, h1i, nullptr, inv_ri, nullptr, nullptr,
        2, 7, (int)NI, 2 * K, Wp2i, b2i, 1.0f, 0, h1i, outI);
  }
}